// InterAggregator_17025250361956
// MI455X (gfx1250) — compile-verified
//
#include <hip/hip_runtime.h>
#include <math.h>

typedef float v2f __attribute__((ext_vector_type(2)));
typedef float v8f __attribute__((ext_vector_type(8)));

#define R_REL 3
#define NPICK 4096
#define KN 32
#define FDIM 256
#define DEMB 128
#define HDIM 16
#define NKV (NPICK*KN)                  /* 131072 */
#define ROWS_PER_REL (NPICK + 2*NKV)    /* 266240 */
#define TILES_PER_REL (ROWS_PER_REL/16) /* 16640  */

/* ---- workspace layout, 32-bit word offsets ---- */
#define OFF_DP   0                /* d(picked)  3*4096            */
#define OFF_DN   12288            /* d(neigh)   3*131072          */
#define OFF_DC   405504           /* d(cand)    3*131072          */
#define OFF_X    798720           /* X = center+agg  4096*256     */
#define OFF_RHOP 1847296          /* rho_plus[3]                  */
#define OFF_RHOM 1847299          /* rho_minus[3]                 */
#define OFF_U64  1847304          /* 8B aligned: suma[3] + subs[3*1024] as u64 fixed point */
#define N_SUMA   3
#define OFF_HIST 1853454          /* int hist    3*1024           */
#define OFF_SUBH 1856526          /* int subhist 3*1024           */
#define OFF_BIN  1859598          /* int bin[3]                   */
#define OFF_NEED 1859601          /* int need[3]                  */
#define WS_WORDS 1859604
#define ZERO_BASE OFF_U64
#define ZERO_WORDS (WS_WORDS - OFF_U64)

__global__ void zero_ws(int* p, int n) {
  int i = blockIdx.x*blockDim.x + threadIdx.x;
  if (i < n) p[i] = 0;
}

/* ------------------------------------------------------------------ */
/* Kernel 1: distance-net scores for picked/neigh/cand rows via WMMA. */
/* One wave = one 16-row tile; layer1 (256->16) uses 64x              */
/* V_WMMA_F32_16X16X4_F32; layer2 (16->1) via shfl_xor reduction.     */
/* ------------------------------------------------------------------ */
__global__ __launch_bounds__(256) void score_kernel(
    const float* __restrict__ feat, const float* __restrict__ W1,
    const float* __restrict__ b1,   const float* __restrict__ W2,
    const float* __restrict__ b2,   const int* __restrict__ picked,
    const int* __restrict__ neigh,  const int* __restrict__ cand,
    float* __restrict__ ws)
{
  __shared__ float W1s[FDIM*HDIM];
  __shared__ float b1s[HDIM];
  __shared__ float w2c[HDIM];
  const int r   = blockIdx.y;
  const int tid = threadIdx.x;
  for (int i = tid; i < FDIM*HDIM; i += 256) W1s[i] = W1[r*FDIM*HDIM + i];
  if (tid < HDIM) { b1s[tid] = b1[r*HDIM + tid]; w2c[tid] = W2[(r*HDIM + tid)*2 + 1]; }
  __syncthreads();

  const int wave = tid >> 5, lane = tid & 31;
  const int tile = blockIdx.x*8 + wave;          /* grid sized exactly */
  const int row0 = tile*16;
  const int m = lane & 15, half = lane >> 4;

  /* A-fragment row -> node index (gather source) */
  int grow = row0 + m;
  int node;
  if      (grow < NPICK)       node = picked[r*NPICK + grow];
  else if (grow < NPICK + NKV) node = neigh [r*NKV + (grow - NPICK)];
  else                         node = cand  [r*NKV + (grow - NPICK - NKV)];

  const v2f* __restrict__ f2 = (const v2f*)feat;
  const long abase = (long)node * (FDIM/2);

  v8f c; for (int i = 0; i < 8; ++i) c[i] = 0.0f;
  #pragma unroll 4
  for (int kk = 0; kk < FDIM/4; ++kk) {
    const int k0 = kk*4 + half*2;
    v2f a = f2[abase + (k0 >> 1)];               /* A[m][k0], A[m][k0+1]   */
    v2f b;
    b.x = W1s[(k0    )*HDIM + m];                /* B[k0][m]               */
    b.y = W1s[(k0 + 1)*HDIM + m];                /* B[k0+1][m]             */
    c = __builtin_amdgcn_wmma_f32_16x16x4_f32(false, a, false, b,
                                              (short)0, c, false, false);
  }

  /* layer2: lane holds hidden j=m, rows M=v+8*half; reduce over j      */
  const float b2v = b2[r*2 + 1];
  #pragma unroll
  for (int v = 0; v < 8; ++v) {
    float h = c[v] + b1s[m];
    h = h > 0.0f ? h : 0.0f;
    float p = h * w2c[m];
    p += __shfl_xor(p, 1, 32);
    p += __shfl_xor(p, 2, 32);
    p += __shfl_xor(p, 4, 32);
    p += __shfl_xor(p, 8, 32);                   /* sum over 16-lane half */
    if (m == v) {                                /* lanes v and 16+v write */
      const int wrow = row0 + v + 8*half;
      const float s = 1.0f / (1.0f + __expf(-(p + b2v)));
      long off;
      if      (wrow < NPICK)       off = OFF_DP + (long)r*NPICK + wrow;
      else if (wrow < NPICK + NKV) off = OFF_DN + (long)r*NKV + (wrow - NPICK);
      else                         off = OFF_DC + (long)r*NKV + (wrow - NPICK - NKV);
      ws[off] = s;
    }
  }
}

/* ------------------------------------------------------------------ */
/* Kernel 2: rho_plus. Top-K pairwise diffs come only from the        */
/* top-32 x bottom-32 masked values (exact reduction of the O(n^2)    */
/* pairwise top-k). One block per relation.                           */
/* ------------------------------------------------------------------ */
__global__ __launch_bounds__(1024) void rho_plus_kernel(
    const int* __restrict__ labels, float* __restrict__ ws)
{
  __shared__ float vt[NPICK];
  __shared__ float vb[NPICK];
  __shared__ float rv[1024];
  __shared__ int   ri[1024];
  __shared__ float tops[32];
  __shared__ float bots[32];
  __shared__ float diffs[1024];
  __shared__ int   s_npos;
  const int r = blockIdx.x;
  const int tid = threadIdx.x;
  const float* dp = ws + OFF_DP + r*NPICK;

  if (tid == 0) s_npos = 0;
  __syncthreads();
  int loc = 0;
  for (int i = tid; i < NPICK; i += 1024) loc += labels[r*NPICK + i];
  atomicAdd(&s_npos, loc);
  __syncthreads();
  const int npos = s_npos;
  const int minority = (npos <= NPICK - npos) ? 1 : 0;

  for (int i = tid; i < NPICK; i += 1024) {
    const float d = dp[i];
    const bool msk = (labels[r*NPICK + i] == minority);
    vt[i] = msk ? d : -1e30f;
    vb[i] = msk ? d :  1e30f;
  }
  __syncthreads();

  for (int t = 0; t < 32; ++t) {                 /* 32 max-extracts */
    float bv = -1e30f; int bi = 0;
    for (int i = tid; i < NPICK; i += 1024) { float x = vt[i]; if (x > bv) { bv = x; bi = i; } }
    rv[tid] = bv; ri[tid] = bi; __syncthreads();
    for (int s = 512; s > 0; s >>= 1) {
      if (tid < s && rv[tid+s] > rv[tid]) { rv[tid] = rv[tid+s]; ri[tid] = ri[tid+s]; }
      __syncthreads();
    }
    if (tid == 0) { tops[t] = rv[0]; vt[ri[0]] = -1e30f; }
    __syncthreads();
  }
  for (int t = 0; t < 32; ++t) {                 /* 32 min-extracts */
    float bv = 1e30f; int bi = 0;
    for (int i = tid; i < NPICK; i += 1024) { float x = vb[i]; if (x < bv) { bv = x; bi = i; } }
    rv[tid] = bv; ri[tid] = bi; __syncthreads();
    for (int s = 512; s > 0; s >>= 1) {
      if (tid < s && rv[tid+s] < rv[tid]) { rv[tid] = rv[tid+s]; ri[tid] = ri[tid+s]; }
      __syncthreads();
    }
    if (tid == 0) { bots[t] = rv[0]; vb[ri[0]] = 1e30f; }
    __syncthreads();
  }
  diffs[tid % 1024] = 0.0f;                      /* (all 1024 set below) */
  diffs[tid] = fabsf(tops[tid >> 5] - bots[tid & 31]);
  __syncthreads();
  float acc = 0.0f;
  for (int t = 0; t < 32; ++t) {                 /* top-32 of 1024 cross diffs */
    rv[tid] = diffs[tid]; ri[tid] = tid; __syncthreads();
    for (int s = 512; s > 0; s >>= 1) {
      if (tid < s && rv[tid+s] > rv[tid]) { rv[tid] = rv[tid+s]; ri[tid] = ri[tid+s]; }
      __syncthreads();
    }
    if (tid == 0) { acc += rv[0] > 0.0f ? rv[0] : 0.0f; diffs[ri[0]] = -1e30f; }
    __syncthreads();
  }
  if (tid == 0) ws[OFF_RHOP + r] = acc / 32.0f;
}

/* ------------------- rho_minus: histogram select ------------------- */
__global__ __launch_bounds__(256) void hist_kernel(float* __restrict__ ws)
{
  __shared__ int h[1024];
  const int r = blockIdx.y, tid = threadIdx.x;
  for (int i = tid; i < 1024; i += 256) h[i] = 0;
  __syncthreads();
  const float* dp = ws + OFF_DP + r*NPICK;
  const float* dn = ws + OFF_DN + r*NKV;
  int* hist = (int*)ws + OFF_HIST + r*1024;
  for (long i = (long)blockIdx.x*256 + tid; i < NKV; i += (long)gridDim.x*256) {
    const float diff = fabsf(dp[i >> 5] - dn[i]);
    int b = (int)(diff * 1024.0f); b = b > 1023 ? 1023 : (b < 0 ? 0 : b);
    atomicAdd(&h[b], 1);
  }
  __syncthreads();
  for (int i = tid; i < 1024; i += 256) if (h[i]) atomicAdd(&hist[i], h[i]);
}

__global__ void findbin_kernel(float* __restrict__ ws)
{
  if (threadIdx.x) return;
  const int r = blockIdx.x;
  const int* hist = (const int*)ws + OFF_HIST + r*1024;
  int cum = 0, b;
  for (b = 1023; b >= 0; --b) { cum += hist[b]; if (cum >= NPICK/2) break; }
  if (b < 0) b = 0;
  ((int*)ws + OFF_BIN)[r]  = b;
  ((int*)ws + OFF_NEED)[r] = NPICK/2 - (cum - hist[b]);
}

__global__ __launch_bounds__(256) void boundary_kernel(float* __restrict__ ws)
{
  const int r = blockIdx.y, tid = threadIdx.x;
  const int bin = ((const int*)ws + OFF_BIN)[r];
  const float blo = (float)bin * (1.0f/1024.0f);
  const float* dp = ws + OFF_DP + r*NPICK;
  const float* dn = ws + OFF_DN + r*NKV;
  unsigned long long* u64b = (unsigned long long*)((int*)ws + OFF_U64);
  unsigned long long* suma = u64b + r;
  unsigned long long* subs = u64b + N_SUMA + r*1024;
  int* subh = (int*)ws + OFF_SUBH + r*1024;
  for (long i = (long)blockIdx.x*256 + tid; i < NKV; i += (long)gridDim.x*256) {
    const float diff = fabsf(dp[i >> 5] - dn[i]);
    int b = (int)(diff * 1024.0f); b = b > 1023 ? 1023 : (b < 0 ? 0 : b);
    if (b > bin) {                               /* fixed-point u64: deterministic */
      atomicAdd(suma, (unsigned long long)(diff * 4294967296.0f));
    } else if (b == bin) {
      int sb = (int)((diff - blo) * (1024.0f*1024.0f));
      sb = sb > 1023 ? 1023 : (sb < 0 ? 0 : sb);
      atomicAdd(&subh[sb], 1);
      atomicAdd(&subs[sb], (unsigned long long)(diff * 4294967296.0f));
    }
  }
}

__global__ void rho_minus_kernel(float* __restrict__ ws)
{
  if (threadIdx.x) return;
  const int r = blockIdx.x;
  const unsigned long long* u64b = (const unsigned long long*)((int*)ws + OFF_U64);
  const unsigned long long* subs = u64b + N_SUMA + r*1024;
  const int* subh = (const int*)ws + OFF_SUBH + r*1024;
  const int need = ((const int*)ws + OFF_NEED)[r];
  double acc = (double)u64b[r];
  int cnt = 0;
  for (int sb = 1023; sb >= 0 && cnt < need; --sb) {
    const int c = subh[sb];
    if (!c) continue;
    if (cnt + c <= need) { acc += (double)subs[sb]; cnt += c; }
    else { acc += (double)subs[sb] * ((double)(need - cnt) / (double)c); cnt = need; }
  }
  ws[OFF_RHOM + r] = (float)(acc / 4294967296.0 / (double)(NPICK/2));
}

/* ------------------------------------------------------------------ */
/* Kernel: X[node][f] = center + masked neighbor/candidate sums.      */
/* One block per node, threads over feature dim (coalesced; features  */
/* table is L2-resident at 51 MB << 192 MB).                          */
/* ------------------------------------------------------------------ */
__global__ __launch_bounds__(256) void agg_kernel(
    const float* __restrict__ feat, const int* __restrict__ picked,
    const int* __restrict__ neigh,  const int* __restrict__ cand,
    float* __restrict__ ws)
{
  const int node = blockIdx.x, f = threadIdx.x;
  float acc = feat[(long)picked[node]*FDIM + f];     /* picked_nodes[0] row */
  for (int r = 0; r < R_REL; ++r) {
    const float dv = ws[OFF_DP + r*NPICK + node];
    const float rm = ws[OFF_RHOM + r];
    const float rp = ws[OFF_RHOP + r];
    const long base = (long)r*NKV + (long)node*KN;
    for (int k = 0; k < KN; ++k) {
      const float dnb = ws[OFF_DN + base + k];
      if (fabsf(dv - dnb) < rm) acc += feat[(long)neigh[base + k]*FDIM + f];
    }
    for (int k = 0; k < KN; ++k) {
      const float dcd = ws[OFF_DC + base + k];
      if (fabsf(dv - dcd) < rp) acc += feat[(long)cand[base + k]*FDIM + f];
    }
  }
  ws[OFF_X + (long)node*FDIM + f] = acc;
}

/* ------------------------------------------------------------------ */
/* Final GEMM: out[128,4096] = relu(W[128,256] @ X^T) via f32 WMMA.   */
/* 8x256 tiles of 16x16, one wave per tile, 64 WMMA (K=4) each.       */
/* ------------------------------------------------------------------ */
__global__ __launch_bounds__(256) void out_gemm_kernel(
    const float* __restrict__ Wt, const float* __restrict__ ws,
    float* __restrict__ out)
{
  const int tid  = threadIdx.x;
  const int wave = tid >> 5, lane = tid & 31;
  const int gid  = blockIdx.x*8 + wave;    /* 0..2047 */
  const int mt   = gid >> 8;               /* 0..7    */
  const int nt   = gid & 255;              /* 0..255  */
  const int m = lane & 15, half = lane >> 4;
  const v2f* W2v = (const v2f*)Wt;
  const v2f* X2  = (const v2f*)(ws + OFF_X);
  const long wbase = (long)(mt*16 + m) * (FDIM/2);
  const long xbase = (long)(nt*16 + m) * (FDIM/2);
  v8f c; for (int i = 0; i < 8; ++i) c[i] = 0.0f;
  #pragma unroll 4
  for (int kk = 0; kk < FDIM/4; ++kk) {
    const int kh = kk*2 + half;            /* (kk*4 + half*2)/2 */
    v2f a = W2v[wbase + kh];               /* A[m][k], A[m][k+1]        */
    v2f b = X2 [xbase + kh];               /* B[k][m] = X[node=m..][k]  */
    c = __builtin_amdgcn_wmma_f32_16x16x4_f32(false, a, false, b,
                                              (short)0, c, false, false);
  }
  #pragma unroll
  for (int v = 0; v < 8; ++v) {
    const int row = mt*16 + v + 8*half;
    const int col = nt*16 + m;
    const float s = c[v] > 0.0f ? c[v] : 0.0f;
    out[(long)row*NPICK + col] = s;
  }
}

extern "C" void kernel_launch(void* const* d_in, const int* in_sizes, int n_in,
                              void* d_out, int out_size, void* d_ws, size_t ws_size,
                              hipStream_t stream)
{
  const float* feat  = (const float*)d_in[0];
  const float* Wt    = (const float*)d_in[1];
  const float* W1    = (const float*)d_in[2];
  const float* b1    = (const float*)d_in[3];
  const float* W2    = (const float*)d_in[4];
  const float* b2    = (const float*)d_in[5];
  const int* picked  = (const int*)d_in[6];
  const int* labels  = (const int*)d_in[7];
  const int* neigh   = (const int*)d_in[8];
  const int* cand    = (const int*)d_in[9];
  float* ws  = (float*)d_ws;
  float* out = (float*)d_out;
  (void)in_sizes; (void)n_in; (void)out_size; (void)ws_size;

  zero_ws<<<(ZERO_WORDS + 255)/256, 256, 0, stream>>>((int*)ws + ZERO_BASE, ZERO_WORDS);
  dim3 gScore(TILES_PER_REL/8, R_REL);               /* 2080 x 3, 8 waves/block */
  score_kernel<<<gScore, 256, 0, stream>>>(feat, W1, b1, W2, b2,
                                           picked, neigh, cand, ws);
  rho_plus_kernel<<<R_REL, 1024, 0, stream>>>(labels, ws);
  dim3 gHist(64, R_REL);
  hist_kernel<<<gHist, 256, 0, stream>>>(ws);
  findbin_kernel<<<R_REL, 32, 0, stream>>>(ws);
  boundary_kernel<<<gHist, 256, 0, stream>>>(ws);
  rho_minus_kernel<<<R_REL, 32, 0, stream>>>(ws);
  agg_kernel<<<NPICK, 256, 0, stream>>>(feat, picked, neigh, cand, ws);
  out_gemm_kernel<<<256, 256, 0, stream>>>(Wt, ws, out);
}